// FullTensorProduct_84937273245793
// MI455X (gfx1250) — compile-verified
//
#include <hip/hip_runtime.h>
#include <hip/hip_bf16.h>

typedef float v2f __attribute__((ext_vector_type(2)));
typedef float v8f __attribute__((ext_vector_type(8)));

#define NPATH 19
#define NCG   1225
#define NSCAL 1584
#define OUTD  2160
#define IN1D  240
#define IN2D  9
#define ROWS  8
#define BLOCK 256

// X(p, l1,l2,l3, off1,mul1,d1, off2,d2, d3, cgoff)
#define PATH_TABLE(X) \
  X( 0, 0,0,0,   0,64,1, 0,1, 1,    0) \
  X( 1, 0,1,1,   0,64,1, 1,3, 3,    1) \
  X( 2, 0,2,2,   0,64,1, 4,5, 5,   10) \
  X( 3, 1,0,1,  64,32,3, 0,1, 3,   35) \
  X( 4, 1,1,0,  64,32,3, 1,3, 1,   44) \
  X( 5, 1,1,1,  64,32,3, 1,3, 3,   53) \
  X( 6, 1,1,2,  64,32,3, 1,3, 5,   80) \
  X( 7, 1,2,1,  64,32,3, 4,5, 3,  125) \
  X( 8, 1,2,2,  64,32,3, 4,5, 5,  170) \
  X( 9, 1,2,3,  64,32,3, 4,5, 7,  245) \
  X(10, 2,0,2, 160,16,5, 0,1, 5,  350) \
  X(11, 2,1,1, 160,16,5, 1,3, 3,  375) \
  X(12, 2,1,2, 160,16,5, 1,3, 5,  420) \
  X(13, 2,1,3, 160,16,5, 1,3, 7,  495) \
  X(14, 2,2,0, 160,16,5, 4,5, 1,  600) \
  X(15, 2,2,1, 160,16,5, 4,5, 3,  625) \
  X(16, 2,2,2, 160,16,5, 4,5, 5,  700) \
  X(17, 2,2,3, 160,16,5, 4,5, 7,  825) \
  X(18, 2,2,4, 160,16,5, 4,5, 9, 1000)

#define F_L1(p,l1,l2,l3,o1,mu,dd1,o2,dd2,dd3,cgo) l1,
#define F_L2(p,l1,l2,l3,o1,mu,dd1,o2,dd2,dd3,cgo) l2,
#define F_L3(p,l1,l2,l3,o1,mu,dd1,o2,dd2,dd3,cgo) l3,
#define F_O1(p,l1,l2,l3,o1,mu,dd1,o2,dd2,dd3,cgo) o1,
#define F_MU(p,l1,l2,l3,o1,mu,dd1,o2,dd2,dd3,cgo) mu,
#define F_D1(p,l1,l2,l3,o1,mu,dd1,o2,dd2,dd3,cgo) dd1,
#define F_O2(p,l1,l2,l3,o1,mu,dd1,o2,dd2,dd3,cgo) o2,
#define F_D2(p,l1,l2,l3,o1,mu,dd1,o2,dd2,dd3,cgo) dd2,
#define F_D3(p,l1,l2,l3,o1,mu,dd1,o2,dd2,dd3,cgo) dd3,
#define F_CG(p,l1,l2,l3,o1,mu,dd1,o2,dd2,dd3,cgo) cgo,

__constant__ int C_l1[NPATH]   = { PATH_TABLE(F_L1) };
__constant__ int C_l2[NPATH]   = { PATH_TABLE(F_L2) };
__constant__ int C_l3[NPATH]   = { PATH_TABLE(F_L3) };
__constant__ int C_off1[NPATH] = { PATH_TABLE(F_O1) };
__constant__ int C_mul[NPATH]  = { PATH_TABLE(F_MU) };
__constant__ int C_d1[NPATH]   = { PATH_TABLE(F_D1) };
__constant__ int C_off2[NPATH] = { PATH_TABLE(F_O2) };
__constant__ int C_d2[NPATH]   = { PATH_TABLE(F_D2) };
__constant__ int C_d3[NPATH]   = { PATH_TABLE(F_D3) };
__constant__ int C_cgoff[NPATH]= { PATH_TABLE(F_CG) };
// e3nn output-irrep sort permutation (stable sort by (l, -p*(-1)^l))
__constant__ int C_inv[NPATH] = {0,4,14,1,3,7,11,5,15,2,6,10,16,8,12,9,13,17,18};

// Tables produced by the init kernel each call (deterministic, capture-safe).
__device__ float    g_cg[NCG];
__device__ unsigned g_meta[NSCAL];

// ---------------- device-side Wigner-3j (replicates reference math, fp64) ----
__device__ __forceinline__ double dfact(int n) {
    double r = 1.0;
    for (int i = 2; i <= n; ++i) r *= (double)i;
    return r;
}
__device__ __forceinline__ int imax2(int a, int b){ return a > b ? a : b; }
__device__ __forceinline__ int imin2(int a, int b){ return a < b ? a : b; }

__device__ double su2cg(int j1, int j2, int j3, int m1, int m2, int m3) {
    if (m3 != m1 + m2) return 0.0;
    int vmin = imax2(imax2(-j1 + j2 + m3, -j1 + m1), 0);
    int vmax = imin2(imin2(j2 + j3 + m1, j3 - j1 + j2), j3 + m3);
    double C = sqrt((2.0 * j3 + 1.0) * dfact(j3 + j1 - j2) * dfact(j3 - j1 + j2) *
                    dfact(j1 + j2 - j3) / dfact(j1 + j2 + j3 + 1)) *
               sqrt(dfact(j3 + m3) * dfact(j3 - m3) * dfact(j1 - m1) * dfact(j1 + m1) *
                    dfact(j2 - m2) * dfact(j2 + m2));
    double S = 0.0;
    for (int v = vmin; v <= vmax; ++v) {
        double sgn = ((v + j2 + m2) & 1) ? -1.0 : 1.0;
        S += sgn * dfact(j2 + j3 + m1 - v) * dfact(j1 - m1 + v) /
             (dfact(v) * dfact(j3 - j1 + j2 - v) * dfact(j3 + m3 - v) * dfact(v + j1 - j2 - m3));
    }
    return C * S;
}

// real<-complex change of basis q (e3nn convention), with (-i)^l phase
__device__ void buildQ(int l, double qr[5][5], double qi[5][5]) {
    for (int i = 0; i < 5; ++i)
        for (int j = 0; j < 5; ++j) { qr[i][j] = 0.0; qi[i][j] = 0.0; }
    const double is2 = 1.0 / sqrt(2.0);
    for (int m = -l; m < 0; ++m) {
        qr[l + m][l - m] = is2;      // q[l+m, l+|m|] = 1/sqrt2
        qi[l + m][l + m] = -is2;     // q[l+m, l-|m|] = -i/sqrt2
    }
    qr[l][l] = 1.0;
    for (int m = 1; m <= l; ++m) {
        double s = (m & 1) ? -1.0 : 1.0;
        qr[l + m][l + m] = s * is2;  // (-1)^m / sqrt2
        qi[l + m][l - m] = s * is2;  // i*(-1)^m / sqrt2
    }
    // multiply by (-i)^l
    double pr, pi;
    switch (l & 3) {
        case 0: pr = 1;  pi = 0;  break;
        case 1: pr = 0;  pi = -1; break;
        case 2: pr = -1; pi = 0;  break;
        default: pr = 0; pi = 1;  break;
    }
    for (int i = 0; i < 5; ++i)
        for (int j = 0; j < 5; ++j) {
            double r = qr[i][j], im = qi[i][j];
            qr[i][j] = r * pr - im * pi;
            qi[i][j] = r * pi + im * pr;
        }
}

__global__ void tp_init_kernel() {
    int t = threadIdx.x;
    if (t < NPATH) {
        int l1 = C_l1[t], l2 = C_l2[t], l3 = C_l3[t];
        int d1 = 2 * l1 + 1, d2 = 2 * l2 + 1, d3 = 2 * l3 + 1;
        double Ccg[5][5][9];
        for (int i = 0; i < d1; ++i)
            for (int k = 0; k < d2; ++k)
                for (int n = 0; n < d3; ++n)
                    Ccg[i][k][n] = su2cg(l1, l2, l3, i - l1, k - l2, n - l3);
        double Q1r[5][5], Q1i[5][5], Q2r[5][5], Q2i[5][5], Q3r[5][5], Q3i[5][5];
        buildQ(l1, Q1r, Q1i); buildQ(l2, Q2r, Q2i); buildQ(l3, Q3r, Q3i);
        double Cr[5][5][9];
        double nrm = 0.0;
        // einsum('ij,kl,nm,ikn->jlm', Q1, Q2, conj(Q3), Ccg), real part
        for (int j = 0; j < d1; ++j)
            for (int lc = 0; lc < d2; ++lc)
                for (int m = 0; m < d3; ++m) {
                    double ar = 0.0;
                    for (int i = 0; i < d1; ++i) {
                        double q1r = Q1r[i][j], q1i = Q1i[i][j];
                        for (int k = 0; k < d2; ++k) {
                            double q2r = Q2r[k][lc], q2i = Q2i[k][lc];
                            double t12r = q1r * q2r - q1i * q2i;
                            double t12i = q1r * q2i + q1i * q2r;
                            for (int n = 0; n < d3; ++n) {
                                double c = Ccg[i][k][n];
                                if (c == 0.0) continue;
                                double q3r = Q3r[n][m], q3i = -Q3i[n][m]; // conj
                                ar += c * (t12r * q3r - t12i * q3i);
                            }
                        }
                    }
                    Cr[j][lc][m] = ar;
                    nrm += ar * ar;
                }
        double scale = sqrt((double)d3) / sqrt(nrm); // 'component' normalization
        int base = C_cgoff[t];
        for (int j = 0; j < d1; ++j)
            for (int lc = 0; lc < d2; ++lc)
                for (int m = 0; m < d3; ++m)
                    g_cg[base + (j * d2 + lc) * d3 + m] = (float)(Cr[j][lc][m] * scale);
    } else if (t == NPATH) {
        // column metadata for the scalar paths (3..18), in output order
        int col = 0, cnt = 0;
        for (int idx = 0; idx < NPATH; ++idx) {
            int p = C_inv[idx];
            int mul = C_mul[p], d3 = 2 * C_l3[p] + 1;
            if (p >= 3) {
                for (int u = 0; u < mul; ++u)
                    for (int k = 0; k < d3; ++k)
                        g_meta[cnt++] = (unsigned)(col + u * d3 + k) |
                                        ((unsigned)p << 12) | ((unsigned)u << 17) |
                                        ((unsigned)k << 23);
            }
            col += mul * d3;
        }
    }
}

// ---------------------------------------------------------------------------
#if defined(__HIP_DEVICE_COMPILE__) && __has_builtin(__builtin_amdgcn_wmma_f32_16x16x4_f32)
#define HAVE_WMMA4 1
#else
#define HAVE_WMMA4 0
#endif

__global__ __launch_bounds__(BLOCK) void tp_main_kernel(
        const float* __restrict__ in1, const float* __restrict__ in2,
        float* __restrict__ out, int nrows) {
    __shared__ float    s_in1[ROWS * IN1D];
    __shared__ float    s_in2[ROWS * 12];
    __shared__ float    s_cg[NCG];
    __shared__ unsigned s_meta[NSCAL];
    __shared__ float    s_wd[ROWS * 576];   // l1=0 outer-product results [row][u(64)][N(9)]

    const int tid = threadIdx.x;
    const long rowBase = (long)blockIdx.x * ROWS;
    const int rcnt = (nrows - rowBase < ROWS) ? (int)(nrows - rowBase) : ROWS;

    for (int i = tid; i < NCG;   i += BLOCK) s_cg[i]   = g_cg[i];
    for (int i = tid; i < NSCAL; i += BLOCK) s_meta[i] = g_meta[i];
    for (int i = tid; i < rcnt * IN1D; i += BLOCK) s_in1[i] = in1[rowBase * IN1D + i];
    for (int i = tid; i < rcnt * IN2D; i += BLOCK) {
        int r = i / IN2D, c = i - r * IN2D;
        s_in2[r * 12 + c] = in2[rowBase * IN2D + i];
    }
    __syncthreads();

    const int wave = tid >> 5, lane = tid & 31, lh = lane & 15;

    // ---- l1=0 block: out = outer(x1[0:64], x2[0:9]) (CG is exactly identity).
    // One wave per row; A(16x4) has only K=0 column, B(4x16) only K=0 row, so
    // only the (lanes0-15, VGPR0) operand slots carry data.
#if HAVE_WMMA4
    {
        const int row = wave;
        float b0 = (lane < 9) ? s_in2[row * 12 + lane] : 0.0f;
        v2f B = { b0, 0.0f };
        for (int t = 0; t < 4; ++t) {
            float a0 = (lane < 16) ? s_in1[row * IN1D + t * 16 + lh] : 0.0f;
            v2f A = { a0, 0.0f };
            v8f C = {};
            v8f D = __builtin_amdgcn_wmma_f32_16x16x4_f32(
                false, A, false, B, (short)0, C, false, false);
            if (lh < 9) {
                int Mb = (lane < 16) ? 0 : 8;
#pragma unroll
                for (int v = 0; v < 8; ++v)
                    s_wd[row * 576 + (t * 16 + Mb + v) * 9 + lh] = D[v];
            }
        }
    }
#else
    for (int i = tid; i < ROWS * 576; i += BLOCK) {
        int row = i / 576, r = i - row * 576, u = r / 9, N = r - u * 9;
        s_wd[i] = s_in1[row * IN1D + u] * s_in2[row * 12 + N];
    }
#endif
    __syncthreads();

    // ---- coalesced writeout of the three l1=0 output chunks
    for (int i = tid; i < rcnt * 64; i += BLOCK) {      // path0 @ col 0: k-dim 1
        int row = i >> 6, u = i & 63;
        out[(rowBase + row) * OUTD + u] = s_wd[row * 576 + u * 9 + 0];
    }
    for (int i = tid; i < rcnt * 192; i += BLOCK) {     // path1 @ col 112: 64u x 3k
        int row = i / 192, j = i - row * 192, u = j / 3, k = j - u * 3;
        out[(rowBase + row) * OUTD + 112 + j] = s_wd[row * 576 + u * 9 + 1 + k];
    }
    for (int i = tid; i < rcnt * 320; i += BLOCK) {     // path2 @ col 688: 64u x 5k
        int row = i / 320, j = i - row * 320, u = j / 5, k = j - u * 5;
        out[(rowBase + row) * OUTD + 688 + j] = s_wd[row * 576 + u * 9 + 4 + k];
    }

    // ---- remaining 16 paths: metadata-driven sparse-CG contraction, coalesced stores
    for (int e = tid; e < rcnt * NSCAL; e += BLOCK) {
        int row = e / NSCAL, s = e - row * NSCAL;
        unsigned m = s_meta[s];
        int col = (int)(m & 0xFFFu);
        int p   = (int)((m >> 12) & 31u);
        int u   = (int)((m >> 17) & 63u);
        int k   = (int)((m >> 23) & 15u);
        int d1 = C_d1[p], d2 = C_d2[p], d3 = C_d3[p];
        const float* x1 = &s_in1[row * IN1D + C_off1[p] + u * d1];
        const float* x2 = &s_in2[row * 12 + C_off2[p]];
        const float* cg = &s_cg[C_cgoff[p] + k];
        float acc = 0.0f;
        for (int i = 0; i < d1; ++i) {
            float a = x1[i];
            for (int j = 0; j < d2; ++j)
                acc += a * x2[j] * cg[(i * d2 + j) * d3];
        }
        out[(rowBase + row) * OUTD + col] = acc;
    }
}

extern "C" void kernel_launch(void* const* d_in, const int* in_sizes, int n_in,
                              void* d_out, int out_size, void* d_ws, size_t ws_size,
                              hipStream_t stream) {
    (void)n_in; (void)d_ws; (void)ws_size; (void)out_size;
    const float* in1 = (const float*)d_in[0];
    const float* in2 = (const float*)d_in[1];
    float* out = (float*)d_out;
    int nrows = in_sizes[0] / IN1D;   // 50000

    // Regenerate constant tables on device every call (deterministic, capture-safe).
    hipLaunchKernelGGL(tp_init_kernel, dim3(1), dim3(32), 0, stream);

    int grid = (nrows + ROWS - 1) / ROWS;
    hipLaunchKernelGGL(tp_main_kernel, dim3(grid), dim3(BLOCK), 0, stream,
                       in1, in2, out, nrows);
}